// VisualMamba_61598420959542
// MI455X (gfx1250) — compile-verified
//
#include <hip/hip_runtime.h>
#include <hip/hip_bf16.h>
#include <math.h>

// ---------------------------------------------------------------------------
// Visual Mamba forward for MI455X (gfx1250), fp32 end-to-end.
// - GEMMs: V_WMMA_F32_16X16X4_F32, one wave per 16x64 C tile (4 accumulators
//   share one A fragment -> 5 b64 loads per 4 WMMAs).
// - Selective scan: per-batch B/C rows staged to LDS via
//   GLOBAL_LOAD_ASYNC_TO_LDS_B128 (ASYNCcnt) when the builtin is available.
// ---------------------------------------------------------------------------

typedef float v2f __attribute__((ext_vector_type(2)));
typedef float v8f __attribute__((ext_vector_type(8)));
typedef int   v4i __attribute__((vector_size(16)));               // b128 payload
typedef __attribute__((address_space(1))) v4i* gv4i_p;            // global int4*
typedef __attribute__((address_space(3))) v4i* lv4i_p;            // LDS int4*

// Model constants (match reference)
#define BATCH   8
#define LTOK    196          // (224/16)^2
#define DMODEL  384
#define DINNER  768
#define DSTATE  16
#define DTRANK  24
#define DBLCOLS 56           // DTRANK + 2*DSTATE
#define MROWS   (BATCH*LTOK) // 1568

__device__ __forceinline__ float silu_f(float x) {
    return x / (1.f + __expf(-x));
}

// ---------------------------------------------------------------------------
// Generic WMMA GEMM:  C[m, n] = act( sum_k A[m*lda + k] * Bw[n*K + k] + bias[n] )
// Bw is (N, K) row-major (torch-style weight).
// Preconditions (all call sites satisfy): M % 16 == 0, K % 4 == 0, lda/K even.
// act: 0 = identity, 1 = softplus.
// One wave per 16x64 output tile (4 WMMA accumulators sharing one A fragment);
// 8 waves (256 threads) per block. Ragged N handled by clamped loads +
// guarded stores, so the K-loop is branch-free with full EXEC (WMMA req).
// ---------------------------------------------------------------------------
__global__ __launch_bounds__(256)
void gemm_nt_wmma(const float* __restrict__ A, int lda,
                  const float* __restrict__ Bw,
                  const float* __restrict__ bias,
                  float* __restrict__ C, int ldc,
                  int M, int N, int K, int act)
{
    const int tilesN = (N + 63) >> 6;         // 64-column quad tiles
    const int wave   = threadIdx.x >> 5;
    const int lane   = threadIdx.x & 31;
    const int tile   = blockIdx.x * 8 + wave;
    const int tm     = tile / tilesN;
    const int tn     = tile - tm * tilesN;
    if (tm * 16 >= M) return;                 // wave-uniform exit (EXEC stays full)

    const int m0   = tm << 4;
    const int n0   = tn << 6;
    const int half = lane >> 4;               // 0: K pair {k,k+1}; 1: {k+2,k+3}
    const int l16  = lane & 15;

    const int mrow = min(m0 + l16, M - 1);    // clamped: loads always in-bounds
    const float* __restrict__ Arow = A + (size_t)mrow * lda + 2 * half;

    const float* __restrict__ Brow[4];
    #pragma unroll
    for (int j = 0; j < 4; ++j)
        Brow[j] = Bw + (size_t)min(n0 + 16 * j + l16, N - 1) * K + 2 * half;

    v8f acc[4] = {v8f{}, v8f{}, v8f{}, v8f{}};
    #pragma unroll 2
    for (int k = 0; k < K; k += 4) {
        const v2f a = *(const v2f*)(Arow + k);       // global_load_b64
        if ((k & 63) == 0) {
            // Speculative, wave-uniform prefetch (global_prefetch_b8);
            // OOB tails are silently dropped (ISA 10.5).
            __builtin_prefetch(Arow + k + 256, 0, 1);
            __builtin_prefetch(Brow[0] + k + 256, 0, 1);
        }
        #pragma unroll
        for (int j = 0; j < 4; ++j) {
            const v2f b = *(const v2f*)(Brow[j] + k); // global_load_b64
            // D = A(16x4 f32) * B(4x16 f32) + C -> v_wmma_f32_16x16x4_f32
            acc[j] = __builtin_amdgcn_wmma_f32_16x16x4_f32(
                /*neg_a=*/false, a, /*neg_b=*/false, b,
                /*c_mod=*/(short)0, acc[j], /*reuse_a=*/false, /*reuse_b=*/false);
        }
    }

    #pragma unroll
    for (int j = 0; j < 4; ++j) {
        const int col = n0 + 16 * j + l16;
        const float bv = (bias != nullptr) ? bias[min(col, N - 1)] : 0.f;
        #pragma unroll
        for (int r = 0; r < 8; ++r) {
            const int row = m0 + r + 8 * half; // ISA C/D layout: VGPR r -> M=r(+8)
            if (row < M && col < N) {
                float v = acc[j][r] + bv;
                if (act == 1)                  // softplus
                    v = (v > 20.f) ? v : log1pf(__expf(v));
                C[(size_t)row * ldc + col] = v;
            }
        }
    }
}

// ---------------------------------------------------------------------------
// Gather img (B,3,224,224) into patch matrix P (MROWS, 768):
//   P[(b*196 + h*14 + w), c*256 + p*16 + q] = img[b, c, 16h+p, 16w+q]
// ---------------------------------------------------------------------------
__global__ __launch_bounds__(256)
void patch_gather(const float* __restrict__ img, float* __restrict__ P)
{
    int idx = blockIdx.x * blockDim.x + threadIdx.x;
    const int total = MROWS * 768;
    if (idx >= total) return;
    int kcol = idx % 768;
    int m    = idx / 768;
    int q = kcol & 15, p = (kcol >> 4) & 15, c = kcol >> 8;
    int w = m % 14, h = (m / 14) % 14, b = m / 196;
    P[idx] = img[(((size_t)b * 3 + c) * 224 + (h * 16 + p)) * 224 + (w * 16 + q)];
}

// ---------------------------------------------------------------------------
// Causal depthwise conv1d (kernel 4) + SiLU on the xc half of xz.
// ---------------------------------------------------------------------------
__global__ __launch_bounds__(256)
void conv_silu(const float* __restrict__ xz,     // (MROWS, 1536); xc at cols [0,768)
               const float* __restrict__ w,      // (768, 4)
               const float* __restrict__ cb,     // (768)
               float* __restrict__ xconv)        // (MROWS, 768)
{
    int idx = blockIdx.x * blockDim.x + threadIdx.x;
    const int total = MROWS * DINNER;
    if (idx >= total) return;
    int d = idx % DINNER;
    int t = idx / DINNER;
    int l = t % LTOK;
    int b = t / LTOK;
    const float* base = xz + (size_t)b * LTOK * 1536;
    float s = cb[d];
    #pragma unroll
    for (int k = 0; k < 4; ++k) {
        int ls = l - 3 + k;
        if (ls >= 0) s += w[d * 4 + k] * base[(size_t)ls * 1536 + d];
    }
    xconv[idx] = silu_f(s);
}

// ---------------------------------------------------------------------------
// Selective scan, fused with  y = (scan_y + D*xconv) * silu(z).
// grid (BATCH, DINNER/256), block 256: one thread owns (b, d) with h[16], A[16]
// in registers. The (LTOK x 56) [dt|B|C] slice for this batch element is staged
// to LDS once via async global->LDS b128 copies (ASYNCcnt path), then all
// 196*16 B/C reads per thread hit LDS.
// ---------------------------------------------------------------------------
__global__ __launch_bounds__(256)
void selective_scan(const float* __restrict__ dtb,   // (MROWS, 768) softplus'd dt
                    const float* __restrict__ dbl,   // (MROWS, 56): [dt|B|C]
                    const float* __restrict__ xconv, // (MROWS, 768)
                    const float* __restrict__ xz,    // z at cols [768,1536)
                    const float* __restrict__ A_log, // (768, 16)
                    const float* __restrict__ Dp,    // (768)
                    float* __restrict__ y)           // (MROWS, 768)
{
    __shared__ __align__(16) float sdbl[LTOK * DBLCOLS];   // 10976 f = 43,904 B
    const int b = blockIdx.x;
    const int d = blockIdx.y * 256 + threadIdx.x;

    // Stage this batch element's [dt|B|C] rows into LDS.
    {
        const float* gsrc = dbl + (size_t)b * LTOK * DBLCOLS;
        const int nvec = (LTOK * DBLCOLS) / 4;             // 2744 x b128
#if __has_builtin(__builtin_amdgcn_global_load_async_to_lds_b128)
        for (int i = threadIdx.x; i < nvec; i += 256) {
            __builtin_amdgcn_global_load_async_to_lds_b128(
                (gv4i_p)(gsrc + 4 * i),          // global src (AS1, int4)
                (lv4i_p)(sdbl + 4 * i),          // LDS dst (AS3, int4)
                /*offset=*/0, /*cpol=*/0);
        }
#if __has_builtin(__builtin_amdgcn_s_wait_asynccnt)
        __builtin_amdgcn_s_wait_asynccnt(0);
#else
        asm volatile("s_wait_asynccnt 0x0" ::: "memory");
#endif
#else
        for (int i = threadIdx.x; i < nvec; i += 256) {
            ((float4*)sdbl)[i] = ((const float4*)gsrc)[i];
        }
#endif
        __syncthreads();
    }

    float A[DSTATE], h[DSTATE];
    #pragma unroll
    for (int n = 0; n < DSTATE; ++n) {
        A[n] = -__expf(A_log[(size_t)d * DSTATE + n]);
        h[n] = 0.f;
    }
    const float Dd = Dp[d];

    for (int l = 0; l < LTOK; ++l) {
        const size_t t = (size_t)b * LTOK + l;
        const float dt = dtb[t * DINNER + d];
        const float xc = xconv[t * DINNER + d];
        const float* bc = sdbl + l * DBLCOLS;  // B at +24, C at +40 (LDS)
        float acc = 0.f;
        #pragma unroll
        for (int n = 0; n < DSTATE; ++n) {
            const float dA = __expf(dt * A[n]);
            h[n] = dA * h[n] + dt * bc[DTRANK + n] * xc;
            acc += h[n] * bc[DTRANK + DSTATE + n];
        }
        const float zv = xz[t * 1536 + DINNER + d];
        y[t * DINNER + d] = (acc + Dd * xc) * silu_f(zv);
    }
}

// ---------------------------------------------------------------------------
// Final LayerNorm(384) per token, then mean over 196 tokens -> out (B, 384).
// ---------------------------------------------------------------------------
__global__ __launch_bounds__(384)
void ln_mean(const float* __restrict__ x,   // (MROWS, 384)
             const float* __restrict__ g,
             const float* __restrict__ be,
             float* __restrict__ out)       // (BATCH, 384)
{
    __shared__ float rs[512];
    __shared__ float rq[512];
    const int b = blockIdx.x;
    const int d = threadIdx.x;
    float acc = 0.f;
    const float gd = g[d], bd = be[d];

    for (int l = 0; l < LTOK; ++l) {
        const float v = x[((size_t)b * LTOK + l) * DMODEL + d];
        rs[d] = v;
        rq[d] = v * v;
        if (d < 128) { rs[384 + d] = 0.f; rq[384 + d] = 0.f; }
        __syncthreads();
        for (int s = 256; s > 0; s >>= 1) {
            if (d < s) { rs[d] += rs[d + s]; rq[d] += rq[d + s]; }
            __syncthreads();
        }
        const float mu  = rs[0] * (1.f / DMODEL);
        const float var = rq[0] * (1.f / DMODEL) - mu * mu;
        acc += (v - mu) * rsqrtf(var + 1e-5f) * gd + bd;
        __syncthreads();   // protect rs/rq before next iteration's writes
    }
    out[(size_t)b * DMODEL + d] = acc * (1.f / LTOK);
}

// ---------------------------------------------------------------------------
// Host launcher
// ---------------------------------------------------------------------------
static inline int gemm_blocks(int M, int N) {
    int tiles = ((M + 15) / 16) * ((N + 63) / 64);
    return (tiles + 7) / 8;
}

extern "C" void kernel_launch(void* const* d_in, const int* in_sizes, int n_in,
                              void* d_out, int out_size, void* d_ws, size_t ws_size,
                              hipStream_t stream) {
    (void)in_sizes; (void)n_in; (void)out_size; (void)ws_size;

    const float* img    = (const float*)d_in[0];
    const float* pe_w   = (const float*)d_in[1];   // (384, 768) flat
    const float* pe_b   = (const float*)d_in[2];
    const float* in_w   = (const float*)d_in[3];   // (6, 1536, 384)
    const float* conv_w = (const float*)d_in[4];   // (6, 768, 4)
    const float* conv_b = (const float*)d_in[5];   // (6, 768)
    const float* xp_w   = (const float*)d_in[6];   // (6, 56, 768)
    const float* dtp_w  = (const float*)d_in[7];   // (6, 768, 24)
    const float* dtp_b  = (const float*)d_in[8];   // (6, 768)
    const float* A_log  = (const float*)d_in[9];   // (6, 768, 16)
    const float* Dp     = (const float*)d_in[10];  // (6, 768)
    const float* out_w  = (const float*)d_in[11];  // (6, 384, 768)
    const float* ln_g   = (const float*)d_in[12];
    const float* ln_b   = (const float*)d_in[13];
    float* out = (float*)d_out;

    // Workspace layout (floats)
    float* ws = (float*)d_ws;
    float* Pmat  = ws;                               // 1568*768
    float* xA    = Pmat  + (size_t)MROWS * 768;      // 1568*384
    float* xB    = xA    + (size_t)MROWS * DMODEL;   // 1568*384
    float* xzbuf = xB    + (size_t)MROWS * DMODEL;   // 1568*1536
    float* xconv = xzbuf + (size_t)MROWS * 1536;     // 1568*768
    float* dbl   = xconv + (size_t)MROWS * DINNER;   // 1568*56
    float* dtbuf = dbl   + (size_t)MROWS * DBLCOLS;  // 1568*768
    float* ybuf  = dtbuf + (size_t)MROWS * DINNER;   // 1568*768

    // 1) Patch embed: gather + WMMA GEMM (+ bias)
    {
        int total = MROWS * 768;
        patch_gather<<<(total + 255) / 256, 256, 0, stream>>>(img, Pmat);
        gemm_nt_wmma<<<gemm_blocks(MROWS, DMODEL), 256, 0, stream>>>(
            Pmat, 768, pe_w, pe_b, xA, DMODEL, MROWS, DMODEL, 768, 0);
    }

    float* xcur = xA;
    float* xnxt = xB;

    for (int i = 0; i < 6; ++i) {
        const float* in_w_i   = in_w   + (size_t)i * 1536 * 384;
        const float* conv_w_i = conv_w + (size_t)i * 768 * 4;
        const float* conv_b_i = conv_b + (size_t)i * 768;
        const float* xp_w_i   = xp_w   + (size_t)i * DBLCOLS * 768;
        const float* dtp_w_i  = dtp_w  + (size_t)i * 768 * DTRANK;
        const float* dtp_b_i  = dtp_b  + (size_t)i * 768;
        const float* A_log_i  = A_log  + (size_t)i * 768 * DSTATE;
        const float* Dp_i     = Dp     + (size_t)i * 768;
        const float* out_w_i  = out_w  + (size_t)i * 384 * 768;

        // in_proj: (1568,384) x (1536,384)^T -> xz (1568,1536)
        gemm_nt_wmma<<<gemm_blocks(MROWS, 1536), 256, 0, stream>>>(
            xcur, DMODEL, in_w_i, nullptr, xzbuf, 1536, MROWS, 1536, DMODEL, 0);

        // causal depthwise conv + SiLU on xc half
        {
            int total = MROWS * DINNER;
            conv_silu<<<(total + 255) / 256, 256, 0, stream>>>(
                xzbuf, conv_w_i, conv_b_i, xconv);
        }

        // x_proj: (1568,768) x (56,768)^T -> dbl (1568,56)
        gemm_nt_wmma<<<gemm_blocks(MROWS, DBLCOLS), 256, 0, stream>>>(
            xconv, DINNER, xp_w_i, nullptr, dbl, DBLCOLS, MROWS, DBLCOLS, DINNER, 0);

        // dt_proj + softplus: (1568,24 of dbl) x (768,24)^T -> dt (1568,768)
        gemm_nt_wmma<<<gemm_blocks(MROWS, DINNER), 256, 0, stream>>>(
            dbl, DBLCOLS, dtp_w_i, dtp_b_i, dtbuf, DINNER, MROWS, DINNER, DTRANK, 1);

        // selective scan fused with (y + D*xc) * silu(z)
        {
            dim3 grid(BATCH, DINNER / 256);
            selective_scan<<<grid, 256, 0, stream>>>(
                dtbuf, dbl, xconv, xzbuf, A_log_i, Dp_i, ybuf);
        }

        // out_proj: (1568,768) x (384,768)^T -> x_next (1568,384)
        gemm_nt_wmma<<<gemm_blocks(MROWS, DMODEL), 256, 0, stream>>>(
            ybuf, DINNER, out_w_i, nullptr, xnxt, DMODEL, MROWS, DMODEL, DINNER, 0);

        float* t = xcur; xcur = xnxt; xnxt = t;
    }

    // Final LayerNorm + token mean
    ln_mean<<<BATCH, 384, 0, stream>>>(xcur, ln_g, ln_b, out);
}